// KNNLookup_62818191671785
// MI455X (gfx1250) — compile-verified
//
#include <hip/hip_runtime.h>
#include <hip/hip_bf16.h>

typedef __attribute__((ext_vector_type(2))) float v2f;
typedef __attribute__((ext_vector_type(8))) float v8f;

#define NROWS   16384
#define CDIM    384
#define NCLS    27
#define TOPKK   10

#define RT      128         // rows per workgroup (8 waves x 16)
#define CTILE   128         // cols per tile iteration
#define NPART   4           // column-range partitions (gridDim.y) -> 512 WGs total
#define KC      32          // K chunk streamed through LDS
#define LDSW    36          // KC + 4 pad: float4-aligned rows, conflict-free frag reads
#define CSW     68          // staging width for one 64-col half of the C tile

// ---------------- workspace layout (in floats) ----------------
#define WS_AOFF   0                                   // Aws   : 16384*384 = 6291456
#define WS_POFF   6291456                             // probs : 16384*27  = 442368
#define WS_TOFF   (6291456 + 442368)                  // topidx: 16384*10 ints = 163840
#define WS_COFF   (6291456 + 442368 + 163840)         // consPart: 64
#define WS_PVOFF  (WS_COFF + 64)                      // partial vals: 16384*4*11 = 720896
#define WS_PIOFF  (WS_PVOFF + 720896)                 // partial idx : 720896 ints

// 1) anchors (b,c,h,w) -> A row-major: row = b*4096 + w*64 + h, col = c
__global__ __launch_bounds__(256) void transpose_anchors(const float* __restrict__ in,
                                                         float* __restrict__ Aws) {
  int idx = blockIdx.x * 256 + threadIdx.x;          // reads coalesced over w
  float v = in[idx];
  int w  = idx & 63;
  int h  = (idx >> 6) & 63;
  int bc = idx >> 12;                                 // b*384 + c
  int c  = bc % 384;
  int b  = bc / 384;
  Aws[((size_t)(b * 4096 + w * 64 + h)) * CDIM + c] = v;
}

// 2) softmax rows of P (prob is (b,27,h,w); row = b*4096 + w*64 + h)
__global__ __launch_bounds__(256) void softmax_rows(const float* __restrict__ prob,
                                                    float* __restrict__ probs) {
  int row = blockIdx.x * 256 + threadIdx.x;
  int b  = row >> 12;
  int wi = (row >> 6) & 63;
  int hi = row & 63;
  const float* base = prob + (size_t)b * NCLS * 4096 + hi * 64 + wi;
  float p[NCLS];
  float mx = -3.0e38f;
#pragma unroll
  for (int j = 0; j < NCLS; ++j) { p[j] = base[(size_t)j * 4096]; mx = fmaxf(mx, p[j]); }
  float s = 0.f;
#pragma unroll
  for (int j = 0; j < NCLS; ++j) { p[j] = expf(p[j] - mx); s += p[j]; }
  float inv = 1.0f / s;
  float* dst = probs + (size_t)row * NCLS;
#pragma unroll
  for (int j = 0; j < NCLS; ++j) dst[j] = p[j] * inv;
}

// 3) fused A*A^T (fp32 WMMA 16x16x4, async global->LDS staging) + per-row top-11
//    over this block's column partition. 128x128 tile per WG; K streamed in 32-chunks.
__global__ __launch_bounds__(256) void gemm_topk(const float* __restrict__ Aws,
                                                 float* __restrict__ pv,
                                                 int* __restrict__ pi) {
  __shared__ float lds[2 * RT * LDSW];                // 9216 floats = 36.9KB
  float* As = lds;                                    // RT x LDSW
  float* Bs = lds + RT * LDSW;                        // RT x LDSW
  float* Cs = lds;                                    // overlay: RT x CSW (8704 floats)

  const int tid  = threadIdx.x;
  const int lane = tid & 31;
  const int wv   = tid >> 5;            // 8 waves, each owns 16 rows of the 128-row block
  const int fm   = lane & 15;           // fragment M (or N) index
  const int fk   = (lane >> 4) << 1;    // fragment K offset {0,2}
  const int kk   = lane >> 4;
  const int r0   = blockIdx.x * RT;
  const int cp   = blockIdx.y;          // column partition: cols [cp*4096, cp*4096+4096)
  const int cstart = cp * (NROWS / NPART);

  // async-copy invariants: element e = tid + i*256 -> r = e>>3 (0..127), c4 = e&7
  unsigned ldsAb = (unsigned)(size_t)As;              // generic->LDS: low 32 bits = LDS offset
  unsigned ldsBb = (unsigned)(size_t)Bs;
  unsigned lA[4], lB[4], gAi[4], gBi[4];
#pragma unroll
  for (int i = 0; i < 4; ++i) {
    int e  = tid + i * 256;
    int r  = e >> 3;
    int c4 = e & 7;
    lA[i]  = ldsAb + (unsigned)(r * (LDSW * 4) + c4 * 16);
    lB[i]  = ldsBb + (unsigned)(r * (LDSW * 4) + c4 * 16);
    gAi[i] = (unsigned)((r0 + r) * (CDIM * 4) + c4 * 16);   // byte offsets into Aws
    gBi[i] = (unsigned)(r * (CDIM * 4) + c4 * 16);
  }

  // sorted top-11 (descending); ascending-index scan + strict '>' == jax top_k ties
  float tv[11]; int ti[11];
#pragma unroll
  for (int s = 0; s < 11; ++s) { tv[s] = -3.0e38f; ti[s] = 0x7fffffff; }

  for (int ct = 0; ct < (NROWS / NPART) / CTILE; ++ct) {
    const int c0 = cstart + ct * CTILE;
    const unsigned cbyte = (unsigned)c0 * (CDIM * 4);

    v8f acc[8];
#pragma unroll
    for (int nt = 0; nt < 8; ++nt) {
      v8f z = {0.f, 0.f, 0.f, 0.f, 0.f, 0.f, 0.f, 0.f};
      acc[nt] = z;
    }

    for (int kc = 0; kc < CDIM / KC; ++kc) {
      const unsigned kbyte = (unsigned)(kc * KC * 4);
      __syncthreads();                   // prior readers of LDS are done
#pragma unroll
      for (int i = 0; i < 4; ++i) {
        unsigned goA = gAi[i] + kbyte;
        unsigned goB = gBi[i] + cbyte + kbyte;
        asm volatile("global_load_async_to_lds_b128 %0, %1, %2 offset:0"
                     :: "v"(lA[i]), "v"(goA), "s"(Aws) : "memory");
        asm volatile("global_load_async_to_lds_b128 %0, %1, %2 offset:0"
                     :: "v"(lB[i]), "v"(goB), "s"(Aws) : "memory");
      }
      asm volatile("s_wait_asynccnt 0x0" ::: "memory");
      __syncthreads();

      const float* ap = &As[(wv * 16 + fm) * LDSW + fk];
#pragma unroll
      for (int kb = 0; kb < KC / 4; ++kb) {             // K step 4
        v2f a = *(const v2f*)(ap + kb * 4);
#pragma unroll
        for (int nt = 0; nt < 8; ++nt) {
          v2f b = *(const v2f*)&Bs[(nt * 16 + fm) * LDSW + kb * 4 + fk];
          acc[nt] = __builtin_amdgcn_wmma_f32_16x16x4_f32(
              false, a, false, b, (short)0, acc[nt], false, false);
        }
      }
    }
    __syncthreads();

    // stage + scan the 128x128 result in two 64-col halves (Cs overlays As/Bs)
#pragma unroll
    for (int h = 0; h < 2; ++h) {
#pragma unroll
      for (int q = 0; q < 4; ++q) {
        // D layout: row = wv*16 + v + 8*kk, col = q*16 + fm
#pragma unroll
        for (int v = 0; v < 8; ++v)
          Cs[(wv * 16 + v + 8 * kk) * CSW + q * 16 + fm] = acc[h * 4 + q][v];
      }
      __syncthreads();
      if (tid < RT) {                     // one thread per row, ascending col index
        const float* crow = &Cs[tid * CSW];
        const int cbase = c0 + h * 64;
        for (int j = 0; j < 64; ++j) {
          float v = crow[j];
          if (v > tv[10]) {
            float cv = v; int ci = cbase + j;
#pragma unroll
            for (int s = 0; s < 11; ++s) {
              if (cv > tv[s]) {
                float t0 = tv[s]; int t1 = ti[s];
                tv[s] = cv; ti[s] = ci;
                cv = t0; ci = t1;
              }
            }
          }
        }
      }
      __syncthreads();
    }
  }

  if (tid < RT) {
    const size_t base = ((size_t)(r0 + tid) * NPART + cp) * 11;
#pragma unroll
    for (int s = 0; s < 11; ++s) { pv[base + s] = tv[s]; pi[base + s] = ti[s]; }
  }
}

// 3b) merge the NPART partial top-11 lists per row -> global ranks 1..10 indices.
//     Explicit (val,idx) comparator reproduces jax.lax.top_k tie-breaking.
__global__ __launch_bounds__(256) void merge_topk(const float* __restrict__ pv,
                                                  const int* __restrict__ pi,
                                                  int* __restrict__ topidx) {
  int row = blockIdx.x * 256 + threadIdx.x;
  float tv[11]; int ti[11];
#pragma unroll
  for (int s = 0; s < 11; ++s) { tv[s] = -3.0e38f; ti[s] = 0x7fffffff; }
#pragma unroll
  for (int p = 0; p < NPART; ++p) {
    const size_t base = ((size_t)row * NPART + p) * 11;
#pragma unroll
    for (int s = 0; s < 11; ++s) {
      float cv = pv[base + s];
      int   ci = pi[base + s];
      if (cv > tv[10] || (cv == tv[10] && ci < ti[10])) {
#pragma unroll
        for (int u = 0; u < 11; ++u) {
          if (cv > tv[u] || (cv == tv[u] && ci < ti[u])) {
            float t0 = tv[u]; int t1 = ti[u];
            tv[u] = cv; ti[u] = ci;
            cv = t0; ci = t1;
          }
        }
      }
    }
  }
#pragma unroll
  for (int s = 1; s < 11; ++s)            // drop rank-0 (self), keep ranks 1..10
    topidx[(size_t)row * TOPKK + (s - 1)] = ti[s];
}

// 4) gather neighbors + reshape-einsum + log loss, deterministic block reduction
__global__ __launch_bounds__(256) void sim_loss(const float* __restrict__ probs,
                                                const int* __restrict__ topidx,
                                                float* __restrict__ consPart) {
  __shared__ float red[256];
  int row = blockIdx.x * 256 + threadIdx.x;
  float p[NCLS];
#pragma unroll
  for (int j = 0; j < NCLS; ++j) p[j] = probs[(size_t)row * NCLS + j];
  float sim[TOPKK];
#pragma unroll
  for (int k = 0; k < TOPKK; ++k) sim[k] = 0.f;
#pragma unroll
  for (int t = 0; t < TOPKK; ++t) {
    int nb = topidx[(size_t)row * TOPKK + t];
    const float* q = probs + (size_t)nb * NCLS;
#pragma unroll
    for (int c = 0; c < NCLS; ++c) {
      // flat index f = t*27+c; reference reshape => sim[f%10] += p[f/10]*q[c]
      const int f = t * NCLS + c;
      sim[f % TOPKK] += p[f / TOPKK] * q[c];
    }
  }
  float contrib = 0.f;
#pragma unroll
  for (int k = 0; k < TOPKK; ++k) contrib += fmaxf(logf(sim[k]), -100.0f);

  red[threadIdx.x] = contrib;
  __syncthreads();
  for (int off = 128; off > 0; off >>= 1) {
    if (threadIdx.x < off) red[threadIdx.x] += red[threadIdx.x + off];
    __syncthreads();
  }
  if (threadIdx.x == 0) consPart[blockIdx.x] = red[0];
}

// 5) class means + entropy + final combine (single block, fully deterministic)
__global__ __launch_bounds__(256) void finalize(const float* __restrict__ probs,
                                                const float* __restrict__ consPart,
                                                float* __restrict__ out) {
  __shared__ float cls[256 * 28];
  int tid = threadIdx.x;
  float acc[NCLS];
#pragma unroll
  for (int j = 0; j < NCLS; ++j) acc[j] = 0.f;
  for (int row = tid; row < NROWS; row += 256) {
#pragma unroll
    for (int j = 0; j < NCLS; ++j) acc[j] += probs[(size_t)row * NCLS + j];
  }
#pragma unroll
  for (int j = 0; j < NCLS; ++j) cls[tid * 28 + j] = acc[j];
  __syncthreads();
  for (int off = 128; off > 0; off >>= 1) {
    if (tid < off)
      for (int j = 0; j < NCLS; ++j) cls[tid * 28 + j] += cls[(tid + off) * 28 + j];
    __syncthreads();
  }
  if (tid == 0) {
    float cs = 0.f;
    for (int i = 0; i < 64; ++i) cs += consPart[i];
    float cons = -cs / ((float)NROWS * (float)TOPKK);
    float ent = 0.f;
    for (int j = 0; j < NCLS; ++j) {
      float m = cls[j] * (1.0f / (float)NROWS);
      m = fmaxf(m, 1e-8f);
      ent -= m * logf(m);
    }
    out[0] = cons - 2.0f * ent;   // total_loss
    out[1] = cons;                // consistency_loss
    out[2] = ent;                 // entropy_loss
  }
}

extern "C" void kernel_launch(void* const* d_in, const int* in_sizes, int n_in,
                              void* d_out, int out_size, void* d_ws, size_t ws_size,
                              hipStream_t stream) {
  const float* anchors = (const float*)d_in[0];   // (4,384,64,64) fp32
  const float* prob    = (const float*)d_in[1];   // (4,27,64,64) fp32
  float* out = (float*)d_out;

  float* ws       = (float*)d_ws;
  float* Aws      = ws + WS_AOFF;
  float* probs    = ws + WS_POFF;
  int*   topidx   = (int*)(ws + WS_TOFF);
  float* consPart = ws + WS_COFF;
  float* pv       = ws + WS_PVOFF;
  int*   pi       = (int*)(ws + WS_PIOFF);

  transpose_anchors<<<NROWS * CDIM / 256, 256, 0, stream>>>(anchors, Aws);
  softmax_rows<<<NROWS / 256, 256, 0, stream>>>(prob, probs);
  gemm_topk<<<dim3(NROWS / RT, NPART), 256, 0, stream>>>(Aws, pv, pi);
  merge_topk<<<NROWS / 256, 256, 0, stream>>>(pv, pi, topidx);
  sim_loss<<<NROWS / 256, 256, 0, stream>>>(probs, topidx, consPart);
  finalize<<<1, 256, 0, stream>>>(probs, consPart, out);
}